// NewLoss_38096359915604
// MI455X (gfx1250) — compile-verified
//
#include <hip/hip_runtime.h>
#include <math.h>

// Problem constants (from reference)
#define BATCH   4096
#define VOCAB   32000
#define VEC4    (VOCAB / 4)     // 8000 float4 per row
#define TPB     320             // 10 wave32s; 8000 / 320 = 25 float4 per thread
#define ITERS   (VEC4 / TPB)    // 25
#define NWAVES  (TPB / 32)      // 10
#define A_COEF  1.0f
#define B_COEF  0.005f

#ifndef __has_builtin
#define __has_builtin(x) 0
#endif

#if __has_builtin(__builtin_amdgcn_global_load_async_to_lds_b128)
#define USE_ASYNC_LDS 1
#else
#define USE_ASYNC_LDS 0
#endif

// Pointer types the async builtin expects (probe-confirmed: param 0 is
// v4i in address_space(1), param 1 is v4i in address_space(3)).
typedef int v4i __attribute__((ext_vector_type(4)));
typedef __attribute__((address_space(1))) v4i* gbl_v4i_ptr;
typedef __attribute__((address_space(3))) v4i* lds_v4i_ptr;

// Async copy 16 bytes global -> LDS (CDNA5 GLOBAL_LOAD_ASYNC_TO_LDS_B128,
// tracked by ASYNCcnt; no VGPR data round-trip).
__device__ __forceinline__ void async_ld16(const float4* g, float4* l) {
#if USE_ASYNC_LDS
  __builtin_amdgcn_global_load_async_to_lds_b128(
      (gbl_v4i_ptr)g, (lds_v4i_ptr)l, /*imm offset*/ 0, /*cpol*/ 0);
#else
  *l = *g;  // fallback: synchronous load + ds_store
#endif
}

// Wait until at most N async ops remain outstanding (compile-time immediate).
template <int N>
__device__ __forceinline__ void wait_async_le() {
#if __has_builtin(__builtin_amdgcn_s_wait_asynccnt)
  __builtin_amdgcn_s_wait_asynccnt(N);
#elif USE_ASYNC_LDS
  asm volatile("s_wait_asynccnt %0" :: "i"(N) : "memory");
#endif
}

// Pass 1, software-pipelined against the in-flight async loads:
// chunk I is consumed under s_wait_asynccnt (ITERS-1-I), so max/sum/sumsq
// math overlaps the remaining HBM->LDS traffic (async dones arrive in order
// per wave, CDNA5 ISA).
template <int I>
struct Pass1 {
  static __device__ __forceinline__ void run(const float4* lds4, int tid,
                                             float& m, float& s, float& ss) {
    wait_async_le<ITERS - 1 - I>();
    const float4 v = lds4[I * TPB + tid];
    m = fmaxf(m, fmaxf(fmaxf(v.x, v.y), fmaxf(v.z, v.w)));
    s += (v.x + v.y) + (v.z + v.w);
    ss = fmaf(v.x, v.x, ss);
    ss = fmaf(v.y, v.y, ss);
    ss = fmaf(v.z, v.z, ss);
    ss = fmaf(v.w, v.w, ss);
    Pass1<I + 1>::run(lds4, tid, m, s, ss);
  }
};
template <>
struct Pass1<ITERS> {
  static __device__ __forceinline__ void run(const float4*, int,
                                             float&, float&, float&) {}
};

// One workgroup per row. Stage the whole 128KB row in LDS via async copies,
// then two LDS passes: (max, sum, sumsq) and sum(exp(x - M)).
__global__ void __launch_bounds__(TPB)
rowloss_kernel(const float* __restrict__ pred,
               const int* __restrict__ labels,
               float* __restrict__ ws) {
  extern __shared__ float4 lds4[];                 // 8000 float4 = 128000 B
  __shared__ float red_m[NWAVES], red_s[NWAVES], red_ss[NWAVES], red_e[NWAVES];
  __shared__ float bcast[3];                       // M, S, SS

  const int tid = threadIdx.x;
  const int row = blockIdx.x;
  const float4* g = (const float4*)(pred + (size_t)row * VOCAB);

  // ---- Stage row: HBM -> LDS, one pass over HBM only ----
#pragma unroll
  for (int i = 0; i < ITERS; ++i) {
    const int idx = i * TPB + tid;
    async_ld16(g + idx, lds4 + idx);
  }

  // ---- Pass 1 (pipelined): per-thread max / sum / sumsq ----
  float m = -INFINITY, s = 0.0f, ss = 0.0f;
  Pass1<0>::run(lds4, tid, m, s, ss);

  // wave32 butterfly reduce
  for (int off = 16; off; off >>= 1) {
    m  = fmaxf(m, __shfl_xor(m, off, 32));
    s  += __shfl_xor(s,  off, 32);
    ss += __shfl_xor(ss, off, 32);
  }
  const int wid = tid >> 5;
  if ((tid & 31) == 0) { red_m[wid] = m; red_s[wid] = s; red_ss[wid] = ss; }
  __syncthreads();  // all waves past their waits -> staged LDS fully visible

  if (tid == 0) {
    float M = red_m[0], S = red_s[0], SS = red_ss[0];
    for (int i = 1; i < NWAVES; ++i) {
      M = fmaxf(M, red_m[i]);
      S += red_s[i];
      SS += red_ss[i];
    }
    bcast[0] = M; bcast[1] = S; bcast[2] = SS;
  }
  __syncthreads();
  const float M = bcast[0];

  // ---- Pass 2: sum exp(x - M) from LDS ----
  float e = 0.0f;
#pragma unroll
  for (int i = 0; i < ITERS; ++i) {
    const float4 v = lds4[i * TPB + tid];
    e += __expf(v.x - M) + __expf(v.y - M) + __expf(v.z - M) + __expf(v.w - M);
  }
  for (int off = 16; off; off >>= 1) e += __shfl_xor(e, off, 32);
  if ((tid & 31) == 0) red_e[wid] = e;
  __syncthreads();

  if (tid == 0) {
    float E = 0.0f;
    for (int i = 0; i < NWAVES; ++i) E += red_e[i];
    const float S = bcast[1], SS = bcast[2];
    const float logZ = M + __logf(E);
    const int lab = labels[row];
    const float t = ((const float*)lds4)[lab];   // target logit from staged row
    const float ce = logZ - t;                   // -(t - logZ)
    const float snt  = S - t;                    // sum of non-target logits
    const float ssnt = fmaf(-t, t, SS);          // sumsq of non-target logits
    const float neg  = ssnt - (snt * snt) * (1.0f / (float)(VOCAB - 1));
    ws[row] = ce * (A_COEF / (float)BATCH) + B_COEF * neg;
  }
}

// Deterministic (fixed-order, no atomics) reduction of 4096 partials.
__global__ void __launch_bounds__(256)
finalize_kernel(const float* __restrict__ ws, float* __restrict__ out) {
  __shared__ double red[8];
  double acc = 0.0;
  for (int i = threadIdx.x; i < BATCH; i += 256) acc += (double)ws[i];
  for (int off = 16; off; off >>= 1) acc += __shfl_xor(acc, off, 32);
  const int wid = threadIdx.x >> 5;
  if ((threadIdx.x & 31) == 0) red[wid] = acc;
  __syncthreads();
  if (threadIdx.x == 0) {
    double t = 0.0;
    for (int i = 0; i < 8; ++i) t += red[i];
    out[0] = (float)t;
  }
}

extern "C" void kernel_launch(void* const* d_in, const int* in_sizes, int n_in,
                              void* d_out, int out_size, void* d_ws, size_t ws_size,
                              hipStream_t stream) {
  const float* pred  = (const float*)d_in[0];
  const int* labels  = (const int*)d_in[1];
  float* ws  = (float*)d_ws;      // BATCH floats of per-row partials
  float* out = (float*)d_out;     // single fp32 scalar

  const size_t shmem = (size_t)VEC4 * sizeof(float4);  // 128000 B row stage
  rowloss_kernel<<<dim3(BATCH), dim3(TPB), shmem, stream>>>(pred, labels, ws);
  finalize_kernel<<<dim3(1), dim3(256), 0, stream>>>(ws, out);
}